// SGC_19181323944516
// MI455X (gfx1250) — compile-verified
//
#include <hip/hip_runtime.h>
#include <hip/hip_bf16.h>

typedef __attribute__((ext_vector_type(2))) float v2f;
typedef __attribute__((ext_vector_type(8))) float v8f;

#define NN 100000
#define NE 400000
#define F  128

// ---------------- degree / norm ----------------
__global__ void k_init_deg(float* deg, int n) {
    int i = blockIdx.x * blockDim.x + threadIdx.x;
    if (i < n) deg[i] = 1.0f;   // self loop
}
__global__ void k_count_deg(const int* __restrict__ dst, float* deg, int ne) {
    int e = blockIdx.x * blockDim.x + threadIdx.x;
    if (e < ne) atomicAdd(&deg[dst[e]], 1.0f);
}
__global__ void k_finalize_norm(float* deg, int n) {
    int i = blockIdx.x * blockDim.x + threadIdx.x;
    if (i < n) deg[i] = rsqrtf(deg[i]);
}

// ---------------- propagation ----------------
// G = norm ⊙ h_in ; out = G   (self-loop contribution)
__global__ void k_scale_init(const float* __restrict__ hin, const float* __restrict__ norm,
                             float* __restrict__ G, float* __restrict__ out) {
    long long t = (long long)blockIdx.x * blockDim.x + threadIdx.x;   // node*32 + f4
    int node = (int)(t >> 5);
    float nm = norm[node];
    float4 v = ((const float4*)hin)[t];
    v.x *= nm; v.y *= nm; v.z *= nm; v.w *= nm;
    ((float4*)G)[t] = v;
    ((float4*)out)[t] = v;
}
// out[dst] += G[src]
__global__ void k_scatter(const int* __restrict__ src, const int* __restrict__ dst,
                          const float* __restrict__ G, float* __restrict__ out) {
    long long t = (long long)blockIdx.x * blockDim.x + threadIdx.x;   // edge*32 + f4
    int e  = (int)(t >> 5);        // wave-uniform -> scalar load
    int f4 = (int)(t & 31);
    int s = src[e], d = dst[e];
    float4 v = ((const float4*)G)[(long long)s * 32 + f4];
    float* o = out + (long long)d * F + f4 * 4;
    atomicAdd(o + 0, v.x); atomicAdd(o + 1, v.y);
    atomicAdd(o + 2, v.z); atomicAdd(o + 3, v.w);
}
// out *= norm
__global__ void k_scale_out(float* __restrict__ out, const float* __restrict__ norm) {
    long long t = (long long)blockIdx.x * blockDim.x + threadIdx.x;
    int node = (int)(t >> 5);
    float nm = norm[node];
    float4 v = ((float4*)out)[t];
    v.x *= nm; v.y *= nm; v.z *= nm; v.w *= nm;
    ((float4*)out)[t] = v;
}

// ---------------- WMMA in-place layer: Zs[64][128] <- act(Zs @ W + b) ----------------
// One 64x132 LDS tile. Each wave (8 per block) owns a 16-col N-slice: preloads its
// 128x16 W slice into 64 VGPRs (32 x v2f B-fragments), accumulates 4 M-tiles of
// 16x16 in registers (4 x v8f), then block-barriers and writes back in place.
// EXEC is all-ones at every WMMA (no divergence on the path into this function).
__device__ __forceinline__ void wmma_layer_inplace(
    float (*Zs)[132], const float* __restrict__ W, const float* __restrict__ bias,
    int wave, int lane, bool relu) {
    const int nb    = wave * 16;
    const int mrow  = lane & 15;                // A row within half / B column
    const int khalf = (lane < 16) ? 0 : 2;      // ISA 16x4 f32 A/B lane split
    const int mbase = (lane < 16) ? 0 : 8;      // C/D layout: VGPR r -> rows r / r+8

    v2f bf[32];                                 // whole K=128 B slice in regs
#pragma unroll
    for (int s = 0; s < 32; ++s) {
        const int kb = 4 * s + khalf;
        bf[s].x = W[kb * F + nb + mrow];
        bf[s].y = W[(kb + 1) * F + nb + mrow];
    }
    const float bs = bias[nb + mrow];

    v8f c[4];
#pragma unroll
    for (int mt = 0; mt < 4; ++mt) {
        v8f acc = {};
#pragma unroll 8
        for (int s = 0; s < 32; ++s) {
            const int kb = 4 * s + khalf;
            v2f a; a.x = Zs[mt * 16 + mrow][kb]; a.y = Zs[mt * 16 + mrow][kb + 1];
            acc = __builtin_amdgcn_wmma_f32_16x16x4_f32(false, a, false, bf[s],
                                                        (short)0, acc, false, false);
        }
        c[mt] = acc;
    }
    __syncthreads();                            // all waves done reading Zs
#pragma unroll
    for (int mt = 0; mt < 4; ++mt)
#pragma unroll
        for (int r = 0; r < 8; ++r) {
            float v = c[mt][r] + bs;
            if (relu) v = fmaxf(v, 0.0f);
            Zs[mt * 16 + mbase + r][nb + mrow] = v;
        }
    __syncthreads();                            // Zs ready for next consumer
}

// ---------------- SGConv final linear: out = H @ W + b (64 rows / block) ----------------
__global__ __launch_bounds__(256) void k_sg_gemm(const float* __restrict__ H,
                                                 const float* __restrict__ W,
                                                 const float* __restrict__ b,
                                                 float* __restrict__ out) {
    __shared__ float Zs[64][132];
    const long long m0 = (long long)blockIdx.x * 64;
    const int t = threadIdx.x;
    // load 64x128 tile (rows clamped for the ragged last block)
#pragma unroll
    for (int pass = 0; pass < 4; ++pass) {
        const int m  = (t >> 4) + 16 * pass;
        const int c0 = t & 15;
        long long row = m0 + m; if (row >= NN) row = NN - 1;
#pragma unroll
        for (int half = 0; half < 2; ++half) {
            const int f4 = c0 + 16 * half;
            float4 v = ((const float4*)H)[row * 32 + f4];
            Zs[m][f4 * 4 + 0] = v.x; Zs[m][f4 * 4 + 1] = v.y;
            Zs[m][f4 * 4 + 2] = v.z; Zs[m][f4 * 4 + 3] = v.w;
        }
    }
    __syncthreads();
    wmma_layer_inplace(Zs, W, b, t >> 5, t & 31, false);
    // writeback (skip OOB rows; divergence is after all WMMAs)
#pragma unroll
    for (int pass = 0; pass < 4; ++pass) {
        const int m  = (t >> 4) + 16 * pass;
        const int c0 = t & 15;
        const long long row = m0 + m;
        if (row < NN) {
#pragma unroll
            for (int half = 0; half < 2; ++half) {
                const int f4 = c0 + 16 * half;
                float4 v = make_float4(Zs[m][f4 * 4 + 0], Zs[m][f4 * 4 + 1],
                                       Zs[m][f4 * 4 + 2], Zs[m][f4 * 4 + 3]);
                ((float4*)out)[row * 32 + f4] = v;
            }
        }
    }
}

// ---------------- fused edge MLP (64 edges / block) ----------------
// z = h[s]⊙h[d]; z = relu(zW1+b1); z = relu(zW2+b2); out = zW3 + b3
__global__ __launch_bounds__(256) void k_edge_mlp(
    const float* __restrict__ H,
    const int* __restrict__ src,  const int* __restrict__ dst,
    const int* __restrict__ nsrc, const int* __restrict__ ndst,
    const float* __restrict__ W1, const float* __restrict__ b1,
    const float* __restrict__ W2, const float* __restrict__ b2,
    const float* __restrict__ W3, const float* __restrict__ b3,
    float* __restrict__ out) {
    __shared__ float Zs[64][132];
    __shared__ int   sidx[64], didx[64];
    __shared__ float red[64];

    const int tilesPerSide = NE / 64;           // 6250
    int tile = blockIdx.x;
    const int* S; const int* D; long long obase;
    if (tile < tilesPerSide) { S = src;  D = dst;  obase = 0; }
    else        { tile -= tilesPerSide; S = nsrc; D = ndst; obase = NE; }
    const int e0 = tile * 64;
    const int t  = threadIdx.x;

    if (t < 64) { sidx[t] = S[e0 + t]; didx[t] = D[e0 + t]; red[t] = 0.0f; }
    __syncthreads();

    // gather + elementwise product: 16 threads per edge row, 2 float4 each, 4 passes
    {
        const float4* Hv = (const float4*)H;
#pragma unroll
        for (int pass = 0; pass < 4; ++pass) {
            const int m  = (t >> 4) + 16 * pass;
            const int c0 = t & 15;
            const long long sb = (long long)sidx[m] * 32;
            const long long db = (long long)didx[m] * 32;
#pragma unroll
            for (int half = 0; half < 2; ++half) {
                const int f4 = c0 + 16 * half;
                float4 a = Hv[sb + f4], b = Hv[db + f4];
                Zs[m][f4 * 4 + 0] = a.x * b.x; Zs[m][f4 * 4 + 1] = a.y * b.y;
                Zs[m][f4 * 4 + 2] = a.z * b.z; Zs[m][f4 * 4 + 3] = a.w * b.w;
            }
        }
    }
    __syncthreads();

    const int wave = t >> 5, lane = t & 31;
    wmma_layer_inplace(Zs, W1, b1, wave, lane, true);
    wmma_layer_inplace(Zs, W2, b2, wave, lane, true);

    // final 128 -> 1 projection: 4 threads per edge, 32 feats each
    {
        const int m  = t >> 2;
        const int c0 = (t & 3) * 32;
        float p = 0.0f;
#pragma unroll
        for (int i = 0; i < 32; ++i) p += Zs[m][c0 + i] * W3[c0 + i];
        atomicAdd(&red[m], p);
    }
    __syncthreads();
    if (t < 64) out[obase + e0 + t] = red[t] + b3[0];
}

// ---------------- launch ----------------
extern "C" void kernel_launch(void* const* d_in, const int* in_sizes, int n_in,
                              void* d_out, int out_size, void* d_ws, size_t ws_size,
                              hipStream_t stream) {
    const float* x    = (const float*)d_in[0];
    const int*   src  = (const int*)  d_in[1];
    const int*   dst  = (const int*)  d_in[2];
    const int*   nsrc = (const int*)  d_in[3];
    const int*   ndst = (const int*)  d_in[4];
    const float* Wsg  = (const float*)d_in[5];
    const float* bsg  = (const float*)d_in[6];
    const float* W1   = (const float*)d_in[7];
    const float* b1   = (const float*)d_in[8];
    const float* W2   = (const float*)d_in[9];
    const float* b2   = (const float*)d_in[10];
    const float* W3   = (const float*)d_in[11];
    const float* b3   = (const float*)d_in[12];
    float* out = (float*)d_out;

    char* ws = (char*)d_ws;
    const unsigned long long HBYTES = (unsigned long long)NN * F * 4ull; // 51.2 MB
    float* norm = (float*)ws;
    float* hA   = (float*)(ws + (1ull << 20));
    float* hB   = (float*)(ws + (1ull << 20) + HBYTES);
    float* hF   = (float*)(ws + (1ull << 20) + 2ull * HBYTES);

    // degrees + norm
    k_init_deg     <<<(NN + 255) / 256, 256, 0, stream>>>(norm, NN);
    k_count_deg    <<<(NE + 255) / 256, 256, 0, stream>>>(dst, norm, NE);
    k_finalize_norm<<<(NN + 255) / 256, 256, 0, stream>>>(norm, NN);

    const int gN = NN * 32 / 256;   // node float4 grid: 12500
    const int gE = NE * 32 / 256;   // edge float4 grid: 50000

    // hop 1: x -> hB  (G scratch = hF)
    k_scale_init<<<gN, 256, 0, stream>>>(x,  norm, hF, hB);
    k_scatter   <<<gE, 256, 0, stream>>>(src, dst, hF, hB);
    k_scale_out <<<gN, 256, 0, stream>>>(hB, norm);
    // hop 2: hB -> hA
    k_scale_init<<<gN, 256, 0, stream>>>(hB, norm, hF, hA);
    k_scatter   <<<gE, 256, 0, stream>>>(src, dst, hF, hA);
    k_scale_out <<<gN, 256, 0, stream>>>(hA, norm);
    // hop 3: hA -> hB
    k_scale_init<<<gN, 256, 0, stream>>>(hA, norm, hF, hB);
    k_scatter   <<<gE, 256, 0, stream>>>(src, dst, hF, hB);
    k_scale_out <<<gN, 256, 0, stream>>>(hB, norm);

    // SGConv linear: hF = hB @ Wsg + bsg   (WMMA f32, 64 rows/block)
    k_sg_gemm<<<(NN + 63) / 64, 256, 0, stream>>>(hB, Wsg, bsg, hF);

    // fused edge MLP (pos + neg), WMMA f32, 64 edges/block
    k_edge_mlp<<<2 * (NE / 64), 256, 0, stream>>>(hF, src, dst, nsrc, ndst,
                                                  W1, b1, W2, b2, W3, b3, out);
}